// myLinear_1_6734508720255
// MI455X (gfx1250) — compile-verified
//
#include <hip/hip_runtime.h>
#include <stdint.h>

// Batched gathered block-GEMM: out[b] = weights[idx[b]] (128x128) @ x[b] (128x32), fp32.
// HBM-bound: 96 MB @ 23.3 TB/s ~= 4.1 us floor; 1.07 GFLOP fp32 via V_WMMA_F32_16X16X4_F32.
// Staging now uses CDNA5 async global->LDS (GLOBAL_LOAD_ASYNC_TO_LDS_B128, ASYNCcnt).

typedef float v2f __attribute__((ext_vector_type(2)));
typedef float v8f __attribute__((ext_vector_type(8)));

#define NB      128   // nodes per block (n)
#define FEAT    32    // feature dim
#define WPAD    132   // padded LDS row stride for W (16B align kept, bank-conflict free)
#define THREADS 256   // 8 wave32 waves

// Low 32 bits of a flat shared-memory pointer == workgroup-relative LDS byte
// address (ISA 10.2: LDS aperture maps addr[31:0] -> LDS offset).
__device__ __forceinline__ unsigned lds_addr_u32(const void* p) {
    return (unsigned)(uintptr_t)p;
}

__device__ __forceinline__ void async_load_b128_to_lds(const void* gaddr, unsigned ldsa) {
    // VDST = per-lane LDS byte address, VADDR = per-lane 64-bit global address,
    // SADDR = off (GV mode). Tracked with ASYNCcnt.
    asm volatile("global_load_async_to_lds_b128 %0, %1, off"
                 :: "v"(ldsa), "v"(gaddr)
                 : "memory");
}

__global__ __launch_bounds__(THREADS) void block_gemm_wmma(
    const float* __restrict__ inp,      // [B*128, 32]
    const float* __restrict__ weights,  // [N_GRAPHS, 128, 128]
    const int*   __restrict__ idx,      // [B]
    float*       __restrict__ out)      // [B*128, 32]
{
    extern __shared__ float smem[];
    float* Wl = smem;                   // 128 * 132 floats
    float* Xl = smem + NB * WPAD;       // 128 * 32 floats

    const int b = blockIdx.x;
    const int t = threadIdx.x;

    const long long g = idx[b];
    const float* Wg = weights + g * (long long)(NB * NB);
    const float* Xg = inp + (long long)b * (NB * FEAT);

    // ---- Stage W (64 KB) and x (16 KB) straight into LDS via async b128 ----
    {
        const float4* Wg4 = (const float4*)Wg;   // 4096 vec4
        #pragma unroll
        for (int j = 0; j < 16; ++j) {
            int i = t + THREADS * j;
            int row = i >> 5;                    // 32 vec4 per 128-float row
            int col = (i & 31) << 2;
            async_load_b128_to_lds(&Wg4[i], lds_addr_u32(&Wl[row * WPAD + col]));
        }
        const float4* Xg4 = (const float4*)Xg;   // 1024 vec4
        #pragma unroll
        for (int j = 0; j < 4; ++j) {
            int i = t + THREADS * j;
            async_load_b128_to_lds(&Xg4[i], lds_addr_u32(&Xl[i << 2]));
        }
    }
    // Wait for this wave's async transfers, then sync all waves' LDS writes.
    asm volatile("s_wait_asynccnt 0x0" ::: "memory");
    __syncthreads();

    // ---- Per-wave WMMA: wave w computes out rows [16w, 16w+16), cols 0..31 ----
    const int wave = t >> 5;        // wave32
    const int lane = t & 31;
    const int half = lane >> 4;     // 0: lanes 0-15, 1: lanes 16-31
    const int ln   = lane & 15;
    const int rowBase = wave * 16;

    v8f acc0 = {};                  // cols 0..15
    v8f acc1 = {};                  // cols 16..31

    // A row pointer for this lane: row = rowBase+ln, K offset 2*half within each step
    const float* Arow = &Wl[(rowBase + ln) * WPAD + 2 * half];

    #pragma unroll 4
    for (int kb = 0; kb < NB; kb += 4) {
        // A fragment (16x4): VGPR0 = K{0|2}, VGPR1 = K{1|3} -> one 8B LDS load
        v2f a = *(const v2f*)&Arow[kb];

        // B fragments (4x16): VGPR0 = row K{0|2}, VGPR1 = row K{1|3}, col = ln (+16)
        int k0 = kb + 2 * half;
        v2f b0, b1;
        b0.x = Xl[(k0    ) * FEAT + ln];
        b0.y = Xl[(k0 + 1) * FEAT + ln];
        b1.x = Xl[(k0    ) * FEAT + ln + 16];
        b1.y = Xl[(k0 + 1) * FEAT + ln + 16];

        // 8 args: (neg_a, A, neg_b, B, c_mod, C, reuse_a, reuse_b)
        acc0 = __builtin_amdgcn_wmma_f32_16x16x4_f32(false, a, false, b0,
                                                     (short)0, acc0, false, false);
        acc1 = __builtin_amdgcn_wmma_f32_16x16x4_f32(false, a, false, b1,
                                                     (short)0, acc1, false, false);
    }

    // ---- Store C/D: VGPR r -> row r (lanes 0-15) / r+8 (lanes 16-31), col = ln ----
    float* Og = out + ((long long)b * NB + rowBase) * FEAT;
    #pragma unroll
    for (int r = 0; r < 8; ++r) {
        int row = r + 8 * half;
        Og[row * FEAT + ln]      = acc0[r];
        Og[row * FEAT + ln + 16] = acc1[r];
    }
}

extern "C" void kernel_launch(void* const* d_in, const int* in_sizes, int n_in,
                              void* d_out, int out_size, void* d_ws, size_t ws_size,
                              hipStream_t stream) {
    const float* inp = (const float*)d_in[0];   // input   [1024*128, 32]
    const float* wts = (const float*)d_in[1];   // weights [1024, 128, 128]
    const int*   idx = (const int*)d_in[2];     // batch_idrange [1024]
    float*       out = (float*)d_out;

    const int B = in_sizes[2];                  // 1024 batch blocks
    const size_t shmem = (size_t)(NB * WPAD + NB * FEAT) * sizeof(float); // ~84 KB

    (void)hipFuncSetAttribute((const void*)block_gemm_wmma,
                              hipFuncAttributeMaxDynamicSharedMemorySize, (int)shmem);

    block_gemm_wmma<<<B, THREADS, shmem, stream>>>(inp, wts, idx, out);
}